// CCPLoss_18872086298967
// MI455X (gfx1250) — compile-verified
//
#include <hip/hip_runtime.h>
#include <stdint.h>

// out = sigmoid( mean | maxpool35(min_C(restored)) - maxpool35(min_C(target)) | )
//
// MI455X schedule: HBM-bound streaming op (100.7 MB inputs; 23.3 TB/s).
// Separable 35-tap max pool as H-pass + V-pass through a 33.5 MB fp32
// intermediate in d_ws => ~168 MB total traffic => ~7.2 us roofline.
// Sliding 35-max via exact radix-5 decomposition (m5 -> m25 -> 25+5+5),
// 5 VALU max ops and 13 LDS loads + 2 LDS stores per element, 3 barrier
// rounds, 2 LDS buffers. The vertical pass stages tiles into LDS with the
// gfx1250 async-copy path (global_load_async_to_lds_b128, ASYNCcnt,
// s_wait_asynccnt). No WMMA: running max is not expressible as A*B+C.
// Deterministic fixed-tree reductions (no float atomics) => bit-stable
// across graph replays.
//
// d_ws layout (requires (2*4194304 + 1024)*4 = 33,558,528 bytes):
//   floats [0, 2*NR)        : hmax intermediate (tensor 0 = restored, 1 = target)
//   floats [2*NR, 2*NR+1024): per-block partial sums

#define NEG_INF (-__builtin_inff())

constexpr int BATCH = 16;
constexpr int CHN   = 3;
constexpr int H     = 512;
constexpr int W     = 512;
constexpr int PADK  = 17;                 // K=35 -> pad 17
constexpr int EXT   = H + 2 * PADK;       // 546 padded length
constexpr long long NR = (long long)BATCH * H * W;   // 4,194,304 per tensor

// ---------------------------------------------------------------------------
// Kernel 1: channel-min + horizontal 35-tap max, one wave (32 lanes) per row.
// grid = 2*16*512/8 = 2048 blocks of 256 threads (8 waves).
// ---------------------------------------------------------------------------
__global__ __launch_bounds__(256) void k1_hpass(const float* __restrict__ restored,
                                                const float* __restrict__ target,
                                                float* __restrict__ hmax) {
  __shared__ float sA[8][EXT + 2];   // E, later overwritten by m25
  __shared__ float sB[8][EXT + 2];   // m5

  const int wave = threadIdx.x >> 5;
  const int lane = threadIdx.x & 31;

  const int rid = blockIdx.x * 8 + wave;     // 0..16383 global row id
  const int t   = rid >> 13;                 // tensor select (8192 rows each)
  const int rem = rid & 8191;
  const int b   = rem >> 9;                  // batch
  const int y   = rem & 511;                 // row

  const float* src = (t ? target : restored) +
                     (((size_t)b * CHN) * H + (size_t)y) * W;
  const float* r0 = src;                 // channel stride = H*W
  const float* r1 = src + (size_t)H * W;
  const float* r2 = src + (size_t)2 * H * W;

  float* A  = sA[wave];
  float* Bv = sB[wave];

  // -inf borders of the extended row E[0..545], payload at [17, 529)
  if (lane < PADK) {
    A[lane]            = NEG_INF;
    A[PADK + W + lane] = NEG_INF;
  }
  // fused channel-min, coalesced 128-bit loads
#pragma unroll
  for (int i = 0; i < 4; ++i) {
    const int x = i * 128 + lane * 4;
    const float4 a = *(const float4*)(r0 + x);
    const float4 c = *(const float4*)(r1 + x);
    const float4 d = *(const float4*)(r2 + x);
    A[PADK + x + 0] = fminf(fminf(a.x, c.x), d.x);
    A[PADK + x + 1] = fminf(fminf(a.y, c.y), d.y);
    A[PADK + x + 2] = fminf(fminf(a.z, c.z), d.z);
    A[PADK + x + 3] = fminf(fminf(a.w, c.w), d.w);
  }
  __syncthreads();

  // radix-5 sliding max: m5[a] = max E[a..a+4]
  for (int i = lane; i < EXT - 4; i += 32)
    Bv[i] = fmaxf(fmaxf(fmaxf(A[i], A[i + 1]), fmaxf(A[i + 2], A[i + 3])),
                  A[i + 4]);
  __syncthreads();
  // m25[a] = max(m5[a], m5[a+5], m5[a+10], m5[a+15], m5[a+20]) -> overwrite E
  for (int i = lane; i < EXT - 24; i += 32)
    A[i] = fmaxf(fmaxf(fmaxf(Bv[i], Bv[i + 5]), fmaxf(Bv[i + 10], Bv[i + 15])),
                 Bv[i + 20]);
  __syncthreads();

  // win35[x] = max(m25[x], m5[x+25], m5[x+30])  (25+5+5 = 35 taps, centered)
  float* out = hmax + (((size_t)(t * BATCH + b)) * H + (size_t)y) * W;
#pragma unroll
  for (int i = 0; i < 4; ++i) {
    const int x = i * 128 + lane * 4;
    float4 o;
    o.x = fmaxf(fmaxf(A[x + 0], Bv[x + 25]), Bv[x + 30]);
    o.y = fmaxf(fmaxf(A[x + 1], Bv[x + 26]), Bv[x + 31]);
    o.z = fmaxf(fmaxf(A[x + 2], Bv[x + 27]), Bv[x + 32]);
    o.w = fmaxf(fmaxf(A[x + 3], Bv[x + 28]), Bv[x + 33]);
    *(float4*)(out + x) = o;
  }
}

// ---------------------------------------------------------------------------
// Kernel 2: vertical 35-tap max on 8-wide column strips + |r-t| accumulate.
// grid = 16 batches * 64 strips = 1024 blocks of 256 threads.
// Tiles staged into LDS with gfx1250 async B128 copies (ASYNCcnt path).
// ---------------------------------------------------------------------------
constexpr int SW = 8;                       // strip width (floats)
constexpr int TILE = EXT * SW;              // 4368 elements

__global__ __launch_bounds__(256) void k2_vpass(const float* __restrict__ hmax,
                                                float* __restrict__ partials) {
  __shared__ __align__(16) float tA[TILE];  // E, later m25
  __shared__ __align__(16) float tB[TILE];  // m5

  const int tid = threadIdx.x;
  const int b   = blockIdx.x >> 6;
  const int x0  = (blockIdx.x & 63) * SW;

  float vr[16];        // restored-pass vertical-max results, register-resident
  float acc = 0.0f;

  for (int tsel = 0; tsel < 2; ++tsel) {
    const float* base = hmax + (size_t)tsel * NR + (size_t)b * H * W;
    const uint64_t sbase = (uint64_t)(uintptr_t)base;

    // Load extended tile E[a][c], a in [0,546), c in [0,8), as 16B chunks:
    // chunk f -> row a = f>>1, half = f&1. Rows are fully valid or fully -inf.
    for (int f = tid; f < EXT * 2; f += 256) {
      const int a = f >> 1;
      const int y = a - PADK;
      if ((unsigned)y < (unsigned)H) {
        const uint32_t voff = (uint32_t)((y * W + x0 + (f & 1) * 4) * 4);
        // generic LDS addresses carry the group-segment offset in [31:0]
        const uint32_t loff = (uint32_t)(uintptr_t)(&tA[f * 4]);
        asm volatile("global_load_async_to_lds_b128 %0, %1, %2"
                     :
                     : "v"(loff), "v"(voff), "s"(sbase)
                     : "memory");
      } else {
        float4 ninf;
        ninf.x = ninf.y = ninf.z = ninf.w = NEG_INF;
        *(float4*)(&tA[f * 4]) = ninf;
      }
    }
    asm volatile("s_wait_asynccnt 0" ::: "memory");
    __syncthreads();

    // m5 (row shift == SW flat elements) -> tB
    for (int i = tid; i < (EXT - 4) * SW; i += 256)
      tB[i] = fmaxf(fmaxf(fmaxf(tA[i], tA[i + SW]),
                          fmaxf(tA[i + 2 * SW], tA[i + 3 * SW])),
                    tA[i + 4 * SW]);
    __syncthreads();
    // m25 -> tA (E dead)
    for (int i = tid; i < (EXT - 24) * SW; i += 256)
      tA[i] = fmaxf(fmaxf(fmaxf(tB[i], tB[i + 5 * SW]),
                          fmaxf(tB[i + 10 * SW], tB[i + 15 * SW])),
                    tB[i + 20 * SW]);
    __syncthreads();

    if (tsel == 0) {
#pragma unroll
      for (int k = 0; k < 16; ++k) {
        const int f = tid + (k << 8);                    // f < 4096 = 512*8
        vr[k] = fmaxf(fmaxf(tA[f], tB[f + 25 * SW]), tB[f + 30 * SW]);
      }
    } else {
#pragma unroll
      for (int k = 0; k < 16; ++k) {
        const int f = tid + (k << 8);
        const float vt = fmaxf(fmaxf(tA[f], tB[f + 25 * SW]), tB[f + 30 * SW]);
        acc += fabsf(vt - vr[k]);
      }
    }
    __syncthreads();   // tiles fully consumed before next phase reuses them
  }

  // deterministic block reduction (fixed tree) into partials[block]
  tA[tid] = acc;
  __syncthreads();
  for (int s = 128; s > 0; s >>= 1) {
    if (tid < s) tA[tid] += tA[tid + s];
    __syncthreads();
  }
  if (tid == 0) partials[blockIdx.x] = tA[0];
}

// ---------------------------------------------------------------------------
// Kernel 3: one wave32 sums 1024 partials deterministically, mean + sigmoid.
// ---------------------------------------------------------------------------
__global__ void k3_final(const float* __restrict__ partials,
                         float* __restrict__ out) {
  float s = 0.0f;
  for (int i = threadIdx.x; i < 1024; i += 32) s += partials[i];
#pragma unroll
  for (int o = 16; o > 0; o >>= 1) s += __shfl_down(s, o, 32);
  if (threadIdx.x == 0) {
    const float mean = s / (float)NR;
    out[0] = 1.0f / (1.0f + __expf(-mean));
  }
}

// ---------------------------------------------------------------------------
extern "C" void kernel_launch(void* const* d_in, const int* in_sizes, int n_in,
                              void* d_out, int out_size, void* d_ws, size_t ws_size,
                              hipStream_t stream) {
  (void)in_sizes; (void)n_in; (void)out_size; (void)ws_size;
  const float* restored = (const float*)d_in[0];
  const float* target   = (const float*)d_in[1];

  float* hmax     = (float*)d_ws;              // 2*NR floats
  float* partials = hmax + (size_t)2 * NR;     // 1024 floats

  k1_hpass<<<2048, 256, 0, stream>>>(restored, target, hmax);
  k2_vpass<<<1024, 256, 0, stream>>>(hmax, partials);
  k3_final<<<1, 32, 0, stream>>>(partials, (float*)d_out);
}